// CWRRTESWindowCell_2001454760114
// MI455X (gfx1250) — compile-verified
//
#include <hip/hip_runtime.h>
#include <hip/hip_bf16.h>
#include <math.h>

typedef float v2f __attribute__((ext_vector_type(2)));
typedef float v8f __attribute__((ext_vector_type(8)));

#define B_   64
#define T_   2048
#define O_   3
#define D_   512
#define H_   4
#define HD_  128
#define M_   100000

// ---------------------------------------------------------------------------
// Workspace layout (bytes):
//   [0,            32768)                : bfrag  float2[128][32]  (B fragments of sal_W)
//   [32768,        32768+2MB)            : logits float[B*T*H]
//   [32768+2MB,    32768+4MB)            : lookup int[B*T*H]
//   [32768+4MB,    32768+6MB)            : weights float[B*T*H]
//   [32768+6MB,    32768+6MB+128KB)      : wvh    float[B*H*HD]
// ---------------------------------------------------------------------------

// K0: pre-swizzle sal_W (512x4, zero-padded to 16 cols) into the 16x16x4 f32
// WMMA B-fragment lane layout: vgpr0 = {K=0 | K=2}, vgpr1 = {K=1 | K=3},
// lanes 0-15 <-> N=0-15 (low K pair), lanes 16-31 <-> N=0-15 (high K pair).
__global__ __launch_bounds__(256) void k_bfrag(const float* __restrict__ salW,
                                               float* __restrict__ bfrag) {
  int e = blockIdx.x * blockDim.x + threadIdx.x;      // 0 .. 4095
  if (e >= 128 * 32) return;
  int kb   = e >> 5;
  int lane = e & 31;
  int n    = lane & 15;
  int k0   = kb * 4 + ((lane < 16) ? 0 : 2);
  float bx = (n < H_) ? salW[k0 * H_ + n]       : 0.f;
  float by = (n < H_) ? salW[(k0 + 1) * H_ + n] : 0.f;
  bfrag[e * 2 + 0] = bx;
  bfrag[e * 2 + 1] = by;
}

// K1: per wave, 16 consecutive positions. Hash -> lookup idx, gather
// engram+embed rows (coalesced 512B b128 loads), fuse x = embed + gate*engram
// into padded LDS, then V_WMMA_F32_16X16X4_F32 K-loop -> logits for 4 heads.
__global__ __launch_bounds__(64) void k_main(
    const int* __restrict__ tokens, const int* __restrict__ prev,
    const float* __restrict__ embed, const float* __restrict__ engram,
    const float* __restrict__ gate_logit, const float* __restrict__ temp,
    const float* __restrict__ sal_b, const float* __restrict__ bfrag,
    float* __restrict__ logits_ws, int* __restrict__ lookup_ws) {
  __shared__ __align__(16) float xs[2][16][132];  // +4 dword row pad -> stride%64==4
  const int lane = threadIdx.x & 31;
  const int w    = threadIdx.x >> 5;
  const int m    = lane & 15;
  const int p0   = blockIdx.x * 32 + w * 16;
  const int b    = p0 / T_;
  const int t0   = p0 - b * T_;
  const int t    = t0 + m;

  // 4-gram context tokens (negative t falls into prev_ids_overlap; PAD==0 so
  // the where(==PAD,0,.) in the reference is the identity)
  int tk[4];
#pragma unroll
  for (int i = 0; i < 4; ++i) {
    int ti = t - i;
    tk[i] = (ti >= 0) ? tokens[b * T_ + ti] : prev[b * O_ + (O_ + ti)];
  }

  v8f c0 = {0.f, 0.f, 0.f, 0.f, 0.f, 0.f, 0.f, 0.f};
  v8f c1 = {0.f, 0.f, 0.f, 0.f, 0.f, 0.f, 0.f, 0.f};
  const float2* __restrict__ bf2 = (const float2*)bfrag;
  const int hiSel = (lane >> 4) << 1;  // 0 for lanes 0-15, 2 for lanes 16-31

#pragma unroll
  for (int h = 0; h < H_; ++h) {
    // rolling-prime hash for this head
    unsigned x = 131u + (unsigned)h * 1009u;
    unsigned hs = 0u;
#pragma unroll
    for (int i = 0; i < 4; ++i) { hs += (unsigned)tk[i] * x; x = x * 31u + 1u; }
    int idx = (int)(hs % (unsigned)M_);
    if (lane < 16) lookup_ws[((size_t)(b * T_ + t)) * H_ + h] = idx;

    // per-lane gate values for this head's 4 columns (reused for 16 rows)
    float4 gl = ((const float4*)gate_logit)[h * 32 + lane];
    float4 g;
    g.x = 1.f / (1.f + __expf(-gl.x));
    g.y = 1.f / (1.f + __expf(-gl.y));
    g.z = 1.f / (1.f + __expf(-gl.z));
    g.w = 1.f / (1.f + __expf(-gl.w));

    // gather: one fully-coalesced 512B row per instruction (32 lanes x b128)
    for (int r = 0; r < 16; ++r) {
      int idxr = __shfl(idx, r, 32);
      int tokr = __shfl(tk[0], r, 32);
      float4 eg = ((const float4*)engram)[((size_t)idxr * H_ + h) * 32 + lane];
      float4 em = ((const float4*)embed)[((size_t)tokr * H_ + h) * 32 + lane];
      float4 v;
      v.x = em.x + g.x * eg.x;
      v.y = em.y + g.y * eg.y;
      v.z = em.z + g.z * eg.z;
      v.w = em.w + g.w * eg.w;
      *(float4*)&xs[w][r][lane * 4] = v;
    }

    // WMMA K-loop over this head's 128 dims (K=4 per op), two accumulators
    // to break the serial C->C chain. A-frag layout: vgpr0={K0|K2},
    // vgpr1={K1|K3}, lanes 0-15/16-31 both map to M=0-15.
#pragma unroll
    for (int j = 0; j < 32; j += 2) {
      v2f a0 = *(const v2f*)&xs[w][m][j * 4 + hiSel];
      float2 bA = bf2[(h * 32 + j) * 32 + lane];
      v2f b0; b0.x = bA.x; b0.y = bA.y;
      c0 = __builtin_amdgcn_wmma_f32_16x16x4_f32(false, a0, false, b0,
                                                 (short)0, c0, false, false);
      v2f a1 = *(const v2f*)&xs[w][m][(j + 1) * 4 + hiSel];
      float2 bB = bf2[(h * 32 + j + 1) * 32 + lane];
      v2f b1; b1.x = bB.x; b1.y = bB.y;
      c1 = __builtin_amdgcn_wmma_f32_16x16x4_f32(false, a1, false, b1,
                                                 (short)0, c1, false, false);
    }
  }

  v8f c = c0 + c1;

  // C layout: element (M=mm, N=n) -> vgpr mm%8, lane n + (mm>=8 ? 16 : 0).
  // Columns n<4 are the head logits.
  int h = lane & 15;
  if (h < H_) {
    float tv   = temp[h];
    float sp   = (tv > 20.f) ? tv : log1pf(__expf(tv));   // softplus
    float invt = 1.f / (sp + 0.3f);
    float bias = sal_b[h];
    int mb = (lane >= 16) ? 8 : 0;
#pragma unroll
    for (int r = 0; r < 8; ++r) {
      float lg = (c[r] + bias) * invt;
      logits_ws[((size_t)(b * T_ + t0 + mb + r)) * H_ + h] = lg;
    }
  }
}

// K2: masked softmax over T for each (b,h); writes weights (exact 0 where masked).
__global__ __launch_bounds__(256) void k_softmax(
    const float* __restrict__ logits_ws, const unsigned char* __restrict__ mask,
    float* __restrict__ weights_ws) {
  const int b = blockIdx.x >> 2, h = blockIdx.x & 3;
  const int tid = threadIdx.x;
  __shared__ float red[256];

  float sl[8];
  bool mk[8];
  float lm = -1e30f;
#pragma unroll
  for (int i = 0; i < 8; ++i) {
    int t = tid + i * 256;
    float v = logits_ws[((size_t)(b * T_ + t)) * H_ + h];
    bool msk = mask[b * T_ + t] != 0;
    sl[i] = msk ? v : -1e9f;
    mk[i] = msk;
    lm = fmaxf(lm, sl[i]);
  }
  red[tid] = lm;
  __syncthreads();
  for (int s = 128; s > 0; s >>= 1) {
    if (tid < s) red[tid] = fmaxf(red[tid], red[tid + s]);
    __syncthreads();
  }
  float rowmax = red[0];
  __syncthreads();

  float e[8];
  float ls = 0.f;
#pragma unroll
  for (int i = 0; i < 8; ++i) {
    e[i] = mk[i] ? __expf(sl[i] - rowmax) : 0.f;
    ls += e[i];
  }
  red[tid] = ls;
  __syncthreads();
  for (int s = 128; s > 0; s >>= 1) {
    if (tid < s) red[tid] += red[tid + s];
    __syncthreads();
  }
  float inv = 1.f / (red[0] + 1e-6f);
#pragma unroll
  for (int i = 0; i < 8; ++i)
    weights_ws[((size_t)(b * T_ + tid + i * 256)) * H_ + h] = e[i] * inv;
}

// K3: write_vec_heads[b,h,d] = sum_t w[b,t,h] * x[b,t,h*HD+d] (x recomputed
// from cached lookup; w==0 positions skipped -> wave-uniform branch).
__global__ __launch_bounds__(512) void k_wvh(
    const int* __restrict__ tokens, const float* __restrict__ embed,
    const float* __restrict__ engram, const float* __restrict__ gate_logit,
    const float* __restrict__ weights_ws, const int* __restrict__ lookup_ws,
    float* __restrict__ wvh) {
  const int b = blockIdx.x >> 2, h = blockIdx.x & 3;
  const int tid = threadIdx.x;
  const int d = tid & 127;
  const int sub = tid >> 7;  // 4 t-interleaved subgroups (wave-uniform)
  float gate = 1.f / (1.f + __expf(-gate_logit[h * HD_ + d]));
  float acc = 0.f;
  for (int t = sub; t < T_; t += 4) {
    size_t o = (size_t)(b * T_ + t);
    float wgt = weights_ws[o * H_ + h];
    if (wgt != 0.f) {
      int idx = lookup_ws[o * H_ + h];
      int tok = tokens[o];
      float eg = engram[((size_t)idx * H_ + h) * HD_ + d];
      float em = embed[(size_t)tok * D_ + h * HD_ + d];
      acc += wgt * (em + gate * eg);
    }
  }
  __shared__ float red[512];
  red[tid] = acc;
  __syncthreads();
  if (tid < 128) {
    float s = red[tid] + red[tid + 128] + red[tid + 256] + red[tid + 384];
    wvh[(size_t)b * D_ + h * HD_ + d] = s;
  }
}

// K4: per-batch tail: window_valid, per-head gate sigmoid, RMS norm, concat.
__global__ __launch_bounds__(128) void k_tail(
    const float* __restrict__ wvh, const unsigned char* __restrict__ mask,
    const float* __restrict__ gate_W, const float* __restrict__ gate_b,
    const float* __restrict__ rms_scale, float* __restrict__ out) {
  const int b = blockIdx.x;
  const int tid = threadIdx.x;
  __shared__ int redi[128];
  __shared__ float redf[128];

  int anym = 0;
  for (int t = tid; t < T_; t += 128) anym |= (mask[b * T_ + t] != 0);
  redi[tid] = anym;
  __syncthreads();
  for (int s = 64; s > 0; s >>= 1) {
    if (tid < s) redi[tid] |= redi[tid + s];
    __syncthreads();
  }
  float valid = redi[0] ? 1.f : 0.f;
  __syncthreads();

  float v[4];
  float sumsq = 0.f;
#pragma unroll
  for (int cc = 0; cc < 4; ++cc) {
    v[cc] = wvh[(size_t)b * D_ + cc * HD_ + tid];
    sumsq += v[cc] * v[cc];
  }
  redf[tid] = sumsq;
  __syncthreads();
  for (int s = 64; s > 0; s >>= 1) {
    if (tid < s) redf[tid] += redf[tid + s];
    __syncthreads();
  }
  float inv_rms = 1.f / sqrtf(redf[0] / (float)D_ + 1e-6f);
  __syncthreads();

  float gw = gate_W[tid];
  float u[4];
#pragma unroll
  for (int cc = 0; cc < 4; ++cc) {  // head cc occupies dims [cc*128, cc*128+128)
    redf[tid] = v[cc] * gw;
    __syncthreads();
    for (int s = 64; s > 0; s >>= 1) {
      if (tid < s) redf[tid] += redf[tid + s];
      __syncthreads();
    }
    float glog = redf[0] + gate_b[0];
    u[cc] = (1.f / (1.f + __expf(-glog))) * valid;
    __syncthreads();
  }
#pragma unroll
  for (int cc = 0; cc < 4; ++cc) {
    out[(size_t)b * 1024 + cc * HD_ + tid] = v[cc] * inv_rms * rms_scale[cc * HD_ + tid];
    out[(size_t)b * 1024 + 512 + cc * HD_ + tid] = u[cc];
  }
}

extern "C" void kernel_launch(void* const* d_in, const int* in_sizes, int n_in,
                              void* d_out, int out_size, void* d_ws, size_t ws_size,
                              hipStream_t stream) {
  const int*           tokens     = (const int*)d_in[0];
  const int*           prev       = (const int*)d_in[1];
  const unsigned char* mask       = (const unsigned char*)d_in[2];  // jax bool = 1 byte
  const float*         embed      = (const float*)d_in[3];
  const float*         engram     = (const float*)d_in[4];
  const float*         gate_logit = (const float*)d_in[5];
  const float*         temp       = (const float*)d_in[6];
  const float*         salW       = (const float*)d_in[7];
  const float*         salb       = (const float*)d_in[8];
  const float*         gateW      = (const float*)d_in[9];
  const float*         gateb      = (const float*)d_in[10];
  const float*         rms_scale  = (const float*)d_in[11];
  float*               out        = (float*)d_out;

  char* ws = (char*)d_ws;
  const size_t SZ_BTH = (size_t)B_ * T_ * H_ * 4;  // 2 MB
  float* bfrag      = (float*)ws;
  float* logits_ws  = (float*)(ws + 32768);
  int*   lookup_ws  = (int*)  (ws + 32768 + SZ_BTH);
  float* weights_ws = (float*)(ws + 32768 + 2 * SZ_BTH);
  float* wvh        = (float*)(ws + 32768 + 3 * SZ_BTH);

  k_bfrag<<<16, 256, 0, stream>>>(salW, bfrag);
  k_main<<<(B_ * T_) / 32, 64, 0, stream>>>(tokens, prev, embed, engram,
                                            gate_logit, temp, salb, bfrag,
                                            logits_ws, lookup_ws);
  k_softmax<<<B_ * H_, 256, 0, stream>>>(logits_ws, mask, weights_ws);
  k_wvh<<<B_ * H_, 512, 0, stream>>>(tokens, embed, engram, gate_logit,
                                     weights_ws, lookup_ws, wvh);
  k_tail<<<B_, 128, 0, stream>>>(wvh, mask, gateW, gateb, rms_scale, out);
}